// Blur_31610959299061
// MI455X (gfx1250) — compile-verified
//
#include <hip/hip_runtime.h>

// Depthwise 4x4 binomial blur, fp32, NCHW (16,256,128,128), pad (2,1)x(2,1).
// Roofline: 512 MB moved @ 23.3 TB/s => ~23 us floor; ~4 FLOP/byte => memory
// bound. CDNA5 path used: GLOBAL_LOAD_ASYNC_TO_LDS_B128 (ASYNCcnt) to stream
// the halo'd tile into LDS, separable [1,3,3,1] filter in VALU with full
// vertical reuse. ROWS=64 tile: halo read overhead = 67/64 = 1.047x.

#define IMG_H   128
#define IMG_W   128
#define ROWS    64               // output rows per block
#define LROWS   (ROWS + 3)       // 67 input rows (y0-2 .. y0+64)
#define LSTRIDE 136              // floats per LDS row (544 B, multiple of 16 B)
#define NCHUNK  (LROWS * 32)     // 67 rows * 32 x 16B chunks = 2144
// LDS row layout (float indices):
//   [2..3]    zeros -> input cols x=-2,-1
//   [4..131]  data  -> input cols x=0..127   (byte offset 16: 16B-aligned)
//   [132..133] zeros-> input cols x=128,129
// reads for column x touch indices x+2 .. x+5 only.

__global__ __launch_bounds__(256)
void blur4x4_binomial_kernel(const float* __restrict__ src,
                             float* __restrict__ dst) {
    __shared__ float lds[LROWS * LSTRIDE];

    const int tid   = threadIdx.x;
    const int ytile = blockIdx.x;            // 0..1
    const int plane = blockIdx.y;            // 0..B*C-1
    const int y0    = ytile * ROWS;

    const float* splane = src + (size_t)plane * (IMG_H * IMG_W);

    // 32-bit LDS byte address of the tile (low 32 bits of a generic pointer
    // into LDS are the LDS offset on gfx1250).
    const unsigned lbase = (unsigned)(uintptr_t)(void*)lds;

    // ---- 1) zero the horizontal pad columns (disjoint from load targets) ----
    for (int i = tid; i < LROWS * 4; i += 256) {
        int r = i >> 2, j = i & 3;            // j: 0,1 -> idx 2,3 ; 2,3 -> idx 132,133
        int idx = (j < 2) ? (2 + j) : (130 + j);
        lds[r * LSTRIDE + idx] = 0.0f;
    }
    // ---- 2) vertical zero padding: only rows 0,1 (top tile) / 66 (bottom) ----
    if (y0 == 0) {
        // rows r=0,1 correspond to gy=-2,-1 : 256 floats, one per thread
        lds[(tid >> 7) * LSTRIDE + 4 + (tid & 127)] = 0.0f;
    }
    if (y0 + ROWS == IMG_H) {
        // row r=66 corresponds to gy=128 : 128 floats
        if (tid < IMG_W) lds[(LROWS - 1) * LSTRIDE + 4 + tid] = 0.0f;
    }

    // ---- 3) async-copy valid rows global -> LDS, 16B per lane-op ----
    // 2144 chunks over 256 threads: 9 guarded issues per thread, one wait.
    #pragma unroll
    for (int k = 0; k < 9; ++k) {
        int i = tid + k * 256;
        if (k < 8 || i < NCHUNK) {            // k<8 always in range (2047<2144)
            int r  = i >> 5;
            int c  = i & 31;
            int gy = y0 - 2 + r;
            if ((unsigned)gy < IMG_H) {
                unsigned goff = (unsigned)(gy * (IMG_W * 4) + c * 16);              // 16B aligned
                unsigned loff = lbase + (unsigned)(r * (LSTRIDE * 4) + 16 + c * 16); // 16B aligned
                asm volatile("global_load_async_to_lds_b128 %0, %1, %2"
                             :: "v"(loff), "v"(goff), "s"(splane)
                             : "memory");
            }
        }
    }

    // wait for this wave's async LDS fills, then workgroup barrier
    asm volatile("s_wait_asynccnt 0" ::: "memory");
    __syncthreads();

    // ---- 4) separable compute: each thread = 1 column x, 32 rows ----
    const int x   = tid & 127;          // column 0..127
    const int ybl = (tid >> 7) * 32;    // 0 or 32: local output row base

    // horizontal pass: h(r) = 1*L[x-2] + 3*L[x-1] + 3*L[x] + 1*L[x+1]
    #define HF(r) ((lds[(r) * LSTRIDE + 2 + x] + lds[(r) * LSTRIDE + 5 + x]) + \
            3.0f * (lds[(r) * LSTRIDE + 3 + x] + lds[(r) * LSTRIDE + 4 + x]))

    float* dcol = dst + (size_t)plane * (IMG_H * IMG_W)
                      + (size_t)(y0 + ybl) * IMG_W + x;

    float h0 = HF(ybl + 0);
    float h1 = HF(ybl + 1);
    float h2 = HF(ybl + 2);
    #pragma unroll
    for (int k = 0; k < 32; ++k) {
        float h3 = HF(ybl + k + 3);
        float v  = (h0 + h3) + 3.0f * (h1 + h2);   // vertical [1,3,3,1]
        dcol[k * IMG_W] = v * (1.0f / 64.0f);
        h0 = h1; h1 = h2; h2 = h3;
    }
    #undef HF
}

extern "C" void kernel_launch(void* const* d_in, const int* in_sizes, int n_in,
                              void* d_out, int out_size, void* d_ws, size_t ws_size,
                              hipStream_t stream) {
    const float* x = (const float*)d_in[0];
    float* out     = (float*)d_out;

    const int planes = in_sizes[0] / (IMG_H * IMG_W);   // B*C = 4096
    dim3 grid(IMG_H / ROWS, planes);                    // (2, 4096)
    blur4x4_binomial_kernel<<<grid, 256, 0, stream>>>(x, out);
}